// SwitchRouter_65687229825653
// MI455X (gfx1250) — compile-verified
//
#include <hip/hip_runtime.h>

typedef __attribute__((ext_vector_type(2))) float v2f;
typedef __attribute__((ext_vector_type(8))) float v8f;

#define N_EXPERTS 8

// ---------------------------------------------------------------------------
// Kernel 0: build zero-padded 16xD weight matrix Wp (experts 8..15 = 0) and
//           zero the cross-block accumulators. Graph-safe per-launch init.
// ---------------------------------------------------------------------------
__global__ void router_prep_kernel(const float* __restrict__ W,
                                   float* __restrict__ Wp, int D,
                                   float* __restrict__ sum_probs,
                                   float* __restrict__ z_sum,
                                   int* __restrict__ counts) {
    const int i = blockIdx.x * blockDim.x + threadIdx.x;
    const int total = 16 * D;
    if (i < total) {
        const int e = i / D;
        Wp[i] = (e < N_EXPERTS) ? W[i] : 0.0f;   // same row stride for e < 8
    }
    if (blockIdx.x == 0) {
        if (threadIdx.x < N_EXPERTS) { sum_probs[threadIdx.x] = 0.0f; counts[threadIdx.x] = 0; }
        if (threadIdx.x == 0) { *z_sum = 0.0f; }
    }
}

// ---------------------------------------------------------------------------
// Kernel 1: logits via V_WMMA_F32_16X16X4_F32.
// Block = 128 threads = 4 waves. Each wave owns TWO 16-token A tiles sharing
// one B fragment per iteration (no masking in the hot loop: Wp is pre-padded).
// Epilogue: all 32 lanes do softmax/argmax/logsumexp (one token per lane).
// ---------------------------------------------------------------------------
__global__ __launch_bounds__(128)
void router_logits_kernel(const float* __restrict__ x,
                          const float* __restrict__ Wp,
                          int S, int D,
                          int*   __restrict__ expert_idx_ws,
                          float* __restrict__ gate_ws,
                          float* __restrict__ sum_probs,
                          float* __restrict__ z_sum,
                          int*   __restrict__ counts) {
    __shared__ float lds_logits[4][2][16][17];   // [wave][tile][token][expert]
    __shared__ float blk_probs[N_EXPERTS];
    __shared__ int   blk_cnt[N_EXPERTS];
    __shared__ float blk_z;

    const int tid  = threadIdx.x;
    const int wave = tid >> 5;
    const int lane = tid & 31;
    const int half = lane >> 4;        // 0: K pair {0,1}, 1: K pair {2,3}
    const int l16  = lane & 15;

    if (tid < N_EXPERTS) { blk_probs[tid] = 0.0f; blk_cnt[tid] = 0; }
    if (tid == 0) blk_z = 0.0f;
    __syncthreads();

    const int tokBase = blockIdx.x * 128 + wave * 32;
    const float* a0p = x  + (size_t)(tokBase +      l16) * D + 2 * half;
    const float* a1p = x  + (size_t)(tokBase + 16 + l16) * D + 2 * half;
    const float* bp  = Wp + (size_t)l16 * D + 2 * half;   // pre-padded, no mask

    v8f c0 = {0.f, 0.f, 0.f, 0.f, 0.f, 0.f, 0.f, 0.f};
    v8f c1 = {0.f, 0.f, 0.f, 0.f, 0.f, 0.f, 0.f, 0.f};
    for (int k = 0; k < D; k += 4) {
        v2f b  = *(const v2f*)(bp  + k);
        v2f a0 = *(const v2f*)(a0p + k);
        v2f a1 = *(const v2f*)(a1p + k);
        c0 = __builtin_amdgcn_wmma_f32_16x16x4_f32(
                 false, a0, false, b, (short)0, c0, false, false);
        c1 = __builtin_amdgcn_wmma_f32_16x16x4_f32(
                 false, a1, false, b, (short)0, c1, false, false);
    }

    // Spill C tiles: VGPR j holds row M=j (lanes 0-15) / M=j+8 (lanes 16-31).
    #pragma unroll
    for (int j = 0; j < 8; ++j) {
        lds_logits[wave][0][j + half * 8][l16] = c0[j];
        lds_logits[wave][1][j + half * 8][l16] = c1[j];
    }
    __syncthreads();

    // All 32 lanes active: lane<16 -> tile 0 tokens, lane>=16 -> tile 1 tokens.
    {
        const int tok = tokBase + half * 16 + l16;
        float l[N_EXPERTS];
        float m  = -INFINITY;
        int   am = 0;
        #pragma unroll
        for (int e = 0; e < N_EXPERTS; ++e) {
            l[e] = lds_logits[wave][half][l16][e];
            if (l[e] > m) { m = l[e]; am = e; }   // strict > : first max wins
        }
        float denom = 0.0f;
        float pr[N_EXPERTS];
        #pragma unroll
        for (int e = 0; e < N_EXPERTS; ++e) {
            pr[e] = __expf(l[e] - m);
            denom += pr[e];
        }
        const float inv  = 1.0f / denom;
        const float gate = inv;                    // exp(m-m)/denom
        const float lse  = m + __logf(denom);

        expert_idx_ws[tok] = am;
        gate_ws[tok]       = gate;

        atomicAdd(&blk_z, lse * lse);
        atomicAdd(&blk_cnt[am], 1);
        #pragma unroll
        for (int e = 0; e < N_EXPERTS; ++e)
            atomicAdd(&blk_probs[e], pr[e] * inv);
    }
    __syncthreads();

    if (tid < N_EXPERTS) {
        atomicAdd(&sum_probs[tid], blk_probs[tid]);
        atomicAdd(&counts[tid], blk_cnt[tid]);
    }
    if (tid == 0) atomicAdd(z_sum, blk_z);
}

// ---------------------------------------------------------------------------
// Kernel 2: token-order prefix count per expert (capacity gating) + aux loss.
// Single block of 1024 threads; 8-vector Hillis-Steele scan in LDS per tile.
// ---------------------------------------------------------------------------
__global__ __launch_bounds__(1024)
void router_rank_kernel(const int*   __restrict__ expert_idx_ws,
                        const float* __restrict__ gate_ws,
                        const float* __restrict__ sum_probs,
                        const float* __restrict__ z_sum,
                        const int*   __restrict__ counts,
                        float* __restrict__ out,
                        int S, int capacity) {
    __shared__ int scan[1024][N_EXPERTS];   // 32 KB
    __shared__ int base[N_EXPERTS];

    const int t = threadIdx.x;
    if (t < N_EXPERTS) base[t] = 0;
    __syncthreads();

    const int nTiles = S / 1024;
    for (int tile = 0; tile < nTiles; ++tile) {
        const int tok = tile * 1024 + t;
        const int e   = expert_idx_ws[tok];
        #pragma unroll
        for (int j = 0; j < N_EXPERTS; ++j) scan[t][j] = (j == e) ? 1 : 0;
        __syncthreads();

        // Hillis-Steele inclusive scan over 1024 threads, 8 lanes per thread.
        for (int off = 1; off < 1024; off <<= 1) {
            int v[N_EXPERTS];
            if (t >= off) {
                #pragma unroll
                for (int j = 0; j < N_EXPERTS; ++j) v[j] = scan[t - off][j];
            }
            __syncthreads();
            if (t >= off) {
                #pragma unroll
                for (int j = 0; j < N_EXPERTS; ++j) scan[t][j] += v[j];
            }
            __syncthreads();
        }

        const int rank = base[e] + scan[t][e] - 1;     // 0-based token rank
        const float g  = (rank < capacity) ? gate_ws[tok] : 0.0f;
        out[tok]     = (float)e;     // expert_idx, cast to output dtype
        out[S + tok] = g;            // kept_gates
        __syncthreads();

        if (t < N_EXPERTS) base[t] += scan[1023][t];
        __syncthreads();
    }

    if (t == 0) {
        const float fS = (float)S;
        float z  = 0.01f * (*z_sum) / fS;
        float lb = 0.0f;
        #pragma unroll
        for (int j = 0; j < N_EXPERTS; ++j)
            lb += ((float)counts[j] / fS) * (sum_probs[j] / fS);
        lb *= 0.01f * (float)N_EXPERTS;
        out[2 * S] = z + lb;         // aux_loss
    }
}

// ---------------------------------------------------------------------------
extern "C" void kernel_launch(void* const* d_in, const int* in_sizes, int n_in,
                              void* d_out, int out_size, void* d_ws, size_t ws_size,
                              hipStream_t stream) {
    const float* x = (const float*)d_in[0];   // (B,T,D) fp32
    const float* W = (const float*)d_in[1];   // (E,D)   fp32

    const int D = in_sizes[1] / N_EXPERTS;    // 2048
    const int S = in_sizes[0] / D;            // 16384
    // capacity = ceil(S/E * 1.25) = ceil(S*5/32), exact in integers
    const int capacity = (S * 5 + 31) / 32;

    // Workspace layout:
    //   [0, S*4)                    : int   expert_idx
    //   [S*4, S*8)                  : float gate_vals
    //   [S*8, S*8 + 16*D*4)         : float Wp (zero-padded 16 x D weights)
    //   then accumulators (64B-aligned slots)
    char* ws = (char*)d_ws;
    int*    eidx      = (int*)  (ws);
    float*  gate      = (float*)(ws + (size_t)S * 4);
    float*  Wp        = (float*)(ws + (size_t)S * 8);
    size_t  accOff    = (size_t)S * 8 + (size_t)16 * D * 4;
    float*  sum_probs = (float*)(ws + accOff);
    float*  z_sum     = (float*)(ws + accOff + 64);
    int*    counts    = (int*)  (ws + accOff + 128);
    float*  out       = (float*)d_out;

    const int prepTotal  = 16 * D;
    const int prepBlocks = (prepTotal + 255) / 256;
    router_prep_kernel<<<prepBlocks, 256, 0, stream>>>(W, Wp, D, sum_probs, z_sum, counts);

    const int blocks = S / 128;   // 4 waves x 32 tokens per block
    router_logits_kernel<<<blocks, 128, 0, stream>>>(
        x, Wp, S, D, eidx, gate, sum_probs, z_sum, counts);

    router_rank_kernel<<<1, 1024, 0, stream>>>(
        eidx, gate, sum_probs, z_sum, counts, out, S, capacity);
}